// PointCloudLoss_40613210751484
// MI455X (gfx1250) — compile-verified
//
#include <hip/hip_runtime.h>

#define NPTS   2048
#define NB     16
#define GSPLIT 8                    // blocks cooperating per batch
#define NWAVES 16
#define THREADS (NWAVES * 32)       // 512 threads = 16 wave32
#define NTILES (NPTS / 16)          // 128
#define ITERS  50

#define EPS_F    0.0025f                  // blur^2 = 0.05^2
#define INV_EPS  400.0f
// log2-domain constants: t2 = t * log2(e)
#define K2       577.07801640f            // INV_EPS * log2(e)
#define LOGW2    (-11.0f)                 // -ln(2048) * log2(e) == -log2(2048)
#define EPS_LN2  0.00173286795139986f     // EPS_F * ln(2)
#define EMD_W    0.5f

typedef float v2f __attribute__((ext_vector_type(2)));
typedef float v8f __attribute__((ext_vector_type(8)));

#if __has_builtin(__builtin_amdgcn_exp2f)
#define EXP2F(x) __builtin_amdgcn_exp2f(x)
#else
#define EXP2F(x) exp2f(x)
#endif

// Branchless A/B fragment build (f32 16x4 / 4x16 WMMA layout, ISA 7.12.2):
// lanes 0-15 hold K={0,1} -> (x,y) pair; lanes 16-31 hold K={2,3} -> (z,0).
// One ds_load_b64 per fragment from pre-packed pair arrays.
__device__ __forceinline__ v2f mk_frag(const float2* Plo, const float2* Phi,
                                       int base, int lane) {
  const float2* p = (lane & 16) ? Phi : Plo;
  float2 v = p[base + (lane & 15)];
  v2f a; a.x = v.x; a.y = v.y;
  return a;
}

__device__ __forceinline__ v8f wmma_dot(v2f a, v2f b) {
  v8f cz = {};
  // D = A x B + 0 : 16x16 tile of dot products, one instruction.
  return __builtin_amdgcn_wmma_f32_16x16x4_f32(false, a, false, b,
                                               (short)0, cz, false, false);
}

// Cross-block barrier for the GSPLIT blocks of one batch (agent scope,
// monotonic phase counter in workspace, zeroed every launch by memset).
__device__ __forceinline__ void group_barrier(unsigned* ctr, unsigned target) {
  __threadfence();
  __syncthreads();
  if (threadIdx.x == 0) {
    __hip_atomic_fetch_add(ctr, 1u, __ATOMIC_ACQ_REL, __HIP_MEMORY_SCOPE_AGENT);
    while (__hip_atomic_load(ctr, __ATOMIC_ACQUIRE, __HIP_MEMORY_SCOPE_AGENT) < target) {
      __builtin_amdgcn_s_sleep(2);
    }
  }
  __syncthreads();
  __threadfence();
}

// One Sinkhorn half-iteration (log2 domain) for this wave's 16 outputs j:
//   out[j] = -eps*ln2 * LSE2_i( (pot_i - C_ij)*K2/INV_EPS ... ) with
//   C = max(0.5*D, 0).  Per-i terms are precomputed into LDS once per pass.
__device__ __forceinline__ void lse_pass(
    const float2* Alo, const float2* Ahi,     // i-side packed coords (LDS)
    const float2* Blo, const float2* Bhi,     // j-side packed coords (LDS)
    const float* __restrict__ potI,           // potential over i (global)
    const float* sqI, const float* sqJ,       // |p|^2 arrays (LDS)
    float* uArr, float* tclArr,               // per-pass scratch (LDS)
    float* __restrict__ outJ,                 // potential over j (global)
    int jt, int lane)
{
  // Cooperative per-i precompute:
  //   u_i   = (pot_i - 0.5*sq_i) * K2          (raw-C branch, no logw/sqJ)
  //   tcl_i =  pot_i * K2 + LOGW2              (C clamped to 0 branch)
  for (int p = threadIdx.x; p < NPTS; p += THREADS) {
    float pot = potI[p];
    uArr[p]   = fmaf(sqI[p], -0.5f * K2, pot * K2);
    tclArr[p] = fmaf(pot, K2, LOGW2);
  }
  __syncthreads();

  v2f bf = mk_frag(Blo, Bhi, jt, lane);
  int j = jt + (lane & 15);
  float cj = fmaf(sqJ[j], -0.5f * K2, LOGW2);
  float rm = -__builtin_inff(), rs = 0.0f;
  int hi8 = (lane & 16) ? 8 : 0;

  for (int it = 0; it < NTILES; ++it) {
    int ib = it * 16;
    v2f af = mk_frag(Alo, Ahi, ib, lane);
    int i0 = ib + hi8;
    float4 u0 = *(const float4*)(uArr   + i0);
    float4 u1 = *(const float4*)(uArr   + i0 + 4);
    float4 c0 = *(const float4*)(tclArr + i0);
    float4 c1 = *(const float4*)(tclArr + i0 + 4);
    float uv[8]  = {u0.x, u0.y, u0.z, u0.w, u1.x, u1.y, u1.z, u1.w};
    float cv[8]  = {c0.x, c0.y, c0.z, c0.w, c1.x, c1.y, c1.z, c1.w};

    v8f c = wmma_dot(af, bf);   // c[k] = dot(point i0+k, point j)

    float t[8];
#pragma unroll
    for (int k = 0; k < 8; ++k) {
      float traw = fmaf(c[k], K2, uv[k] + cj);
      t[k] = fminf(traw, cv[k]);
    }
    float lm = t[0];
#pragma unroll
    for (int k = 1; k < 8; ++k) lm = fmaxf(lm, t[k]);
    float s = 0.0f;
#pragma unroll
    for (int k = 0; k < 8; ++k) s += EXP2F(t[k] - lm);
    float nm = fmaxf(rm, lm);
    rs = fmaf(rs, EXP2F(rm - nm), s * EXP2F(lm - nm));
    rm = nm;
  }
  // lanes L and L+16 hold the two halves of output j: merge once.
  float om = __shfl_xor(rm, 16, 32);
  float os = __shfl_xor(rs, 16, 32);
  float nm = fmaxf(rm, om);
  float st = rs * EXP2F(rm - nm) + os * EXP2F(om - nm);
  float res = -EPS_LN2 * (nm + __log2f(st));
  if (lane < 16) outJ[j] = res;
}

// Chamfer direction pass: per-wave sum over its 16 outputs j of min_i D_ij,
// written to a private slot (deterministic, no atomics).
__device__ __forceinline__ void min_pass(
    const float2* Alo, const float2* Ahi,     // i-side (reduced)
    const float2* Blo, const float2* Bhi,     // j-side (output)
    const float* sqI, const float* sqJ,
    float* __restrict__ slot, int jt, int lane)
{
  v2f bf = mk_frag(Blo, Bhi, jt, lane);
  int j = jt + (lane & 15);
  float sj = sqJ[j];
  float rmin = __builtin_inff();
  int hi8 = (lane & 16) ? 8 : 0;

  for (int it = 0; it < NTILES; ++it) {
    int ib = it * 16;
    v2f af = mk_frag(Alo, Ahi, ib, lane);
    int i0 = ib + hi8;
    float4 s0 = *(const float4*)(sqI + i0);
    float4 s1 = *(const float4*)(sqI + i0 + 4);
    float sv[8] = {s0.x, s0.y, s0.z, s0.w, s1.x, s1.y, s1.z, s1.w};

    v8f c = wmma_dot(af, bf);
#pragma unroll
    for (int k = 0; k < 8; ++k) {
      float d = fmaf(-2.0f, c[k], sv[k] + sj);
      rmin = fminf(rmin, fmaxf(d, 0.0f));
    }
  }
  rmin = fminf(rmin, __shfl_xor(rmin, 16, 32));
  float sum = rmin;                        // sum the 16 distinct mins
  sum += __shfl_xor(sum, 8, 32);
  sum += __shfl_xor(sum, 4, 32);
  sum += __shfl_xor(sum, 2, 32);
  sum += __shfl_xor(sum, 1, 32);
  if (lane == 0) *slot = sum;
}

__global__ __launch_bounds__(THREADS)
void pc_sinkhorn_kernel(const float* __restrict__ yt, const float* __restrict__ yp,
                        float* __restrict__ fpot, float* __restrict__ gpot,
                        float* __restrict__ partials, unsigned* __restrict__ ctrs)
{
  // 96 KB LDS: packed WMMA-fragment coord pairs, squared norms, pass scratch.
  __shared__ __align__(16) float2 XL[NPTS], XH[NPTS];   // y_true  (x,y) / (z,0)
  __shared__ __align__(16) float2 YL[NPTS], YH[NPTS];   // y_pred  (x,y) / (z,0)
  __shared__ __align__(16) float  xx[NPTS], yy[NPTS];
  __shared__ __align__(16) float  uArr[NPTS], tclArr[NPTS];

  const int b    = blockIdx.x / GSPLIT;
  const int sub  = blockIdx.x % GSPLIT;
  const int tid  = threadIdx.x;
  const int lane = tid & 31;
  const int wave = tid >> 5;
  const int jt   = (sub * NWAVES + wave) * 16;   // this wave's 16 outputs

  const float* xb = yt + (size_t)b * NPTS * 3;
  const float* yb = yp + (size_t)b * NPTS * 3;
  for (int p = tid; p < NPTS; p += THREADS) {
    float a0 = xb[3*p+0], a1 = xb[3*p+1], a2 = xb[3*p+2];
    XL[p] = make_float2(a0, a1);
    XH[p] = make_float2(a2, 0.0f);
    xx[p] = a0*a0 + a1*a1 + a2*a2;
    float c0 = yb[3*p+0], c1 = yb[3*p+1], c2 = yb[3*p+2];
    YL[p] = make_float2(c0, c1);
    YH[p] = make_float2(c2, 0.0f);
    yy[p] = c0*c0 + c1*c1 + c2*c2;
  }
  __syncthreads();

  float*    fb   = fpot + b * NPTS;
  float*    gb   = gpot + b * NPTS;
  unsigned* ctr  = ctrs + b;
  float*    slot = partials + (size_t)(blockIdx.x * NWAVES + wave) * 2;

  // Chamfer: min over m (y_pred) per n, and min over n per m.
  min_pass(YL, YH, XL, XH, yy, xx, slot + 0, jt, lane);
  min_pass(XL, XH, YL, YH, xx, yy, slot + 1, jt, lane);

  // Sinkhorn: f/g start at 0 (workspace memset each launch).
  unsigned phase = 0;
  for (int itr = 0; itr < ITERS; ++itr) {
    lse_pass(YL, YH, XL, XH, gb, yy, xx, uArr, tclArr, fb, jt, lane);  // f
    group_barrier(ctr, (++phase) * GSPLIT);
    lse_pass(XL, XH, YL, YH, fb, xx, yy, uArr, tclArr, gb, jt, lane);  // g
    group_barrier(ctr, (++phase) * GSPLIT);
  }
}

__global__ void pc_finalize_kernel(const float* __restrict__ partials,
                                   const float* __restrict__ fpot,
                                   const float* __restrict__ gpot,
                                   float* __restrict__ out)
{
  __shared__ float cd_sh;
  if (threadIdx.x == 0) {
    float rsum = 0.0f, csum = 0.0f;
    for (int i = 0; i < NB * GSPLIT * NWAVES; ++i) {
      rsum += partials[2*i+0];
      csum += partials[2*i+1];
    }
    cd_sh = rsum / (float)(NB * NPTS) + csum / (float)(NB * NPTS);
  }
  __syncthreads();
  int t = threadIdx.x;
  if (t < NB) {
    const float* fb = fpot + t * NPTS;
    const float* gb = gpot + t * NPTS;
    float fs = 0.0f, gs = 0.0f;
    for (int p = 0; p < NPTS; ++p) { fs += fb[p]; gs += gb[p]; }
    float emd = (fs + gs) * (1.0f / (float)NPTS);
    out[t] = (1.0f - EMD_W) * cd_sh + EMD_W * emd;
  }
}

extern "C" void kernel_launch(void* const* d_in, const int* in_sizes, int n_in,
                              void* d_out, int out_size, void* d_ws, size_t ws_size,
                              hipStream_t stream) {
  (void)in_sizes; (void)n_in; (void)out_size; (void)ws_size;
  const float* yt = (const float*)d_in[0];
  const float* yp = (const float*)d_in[1];
  float* out = (float*)d_out;

  char* ws = (char*)d_ws;
  const size_t CTR_OFF  = 0;                               // NB counters
  const size_t PART_OFF = 256;                             // NB*GSPLIT*NWAVES*2 floats
  const size_t F_OFF    = PART_OFF + (size_t)NB * GSPLIT * NWAVES * 2 * sizeof(float);
  const size_t G_OFF    = F_OFF + (size_t)NB * NPTS * sizeof(float);
  const size_t TOTAL    = G_OFF + (size_t)NB * NPTS * sizeof(float);

  unsigned* ctrs     = (unsigned*)(ws + CTR_OFF);
  float*    partials = (float*)(ws + PART_OFF);
  float*    fpot     = (float*)(ws + F_OFF);
  float*    gpot     = (float*)(ws + G_OFF);

  // Zero barrier counters, chamfer slots, and initial f/g potentials.
  hipMemsetAsync(d_ws, 0, TOTAL, stream);

  pc_sinkhorn_kernel<<<NB * GSPLIT, THREADS, 0, stream>>>(yt, yp, fpot, gpot,
                                                          partials, ctrs);
  pc_finalize_kernel<<<1, 64, 0, stream>>>(partials, fpot, gpot, out);
}